// SimpleAttLSTM_24902220382993
// MI455X (gfx1250) — compile-verified
//
#include <hip/hip_runtime.h>

// ---------------------------------------------------------------------------
// Fused LSTM(H=4, IN=2) + attention pooling + FC for MI455X (gfx1250).
//
// One wave owns 16 batch elements. Per time step, one v_wmma_f32_16x16x4_f32
// computes gates^T = W_hh(16x4) @ h^T(4x16) + C, with C = x@W_ih^T + b built
// by 16 dual-issued FMAs/lane. PyTorch gate order (i,f,g,o) lands i,f in
// lanes 0-15 and g,o in lanes 16-31; halves exchange activations with
// ds_swizzle SWAPX16. Pass 0 -> h_T (query); pass 1 reruns the recurrence
// with an online softmax so h_seq (128 MiB) is never materialized.
//
// x is [B,T,2] so per-thread reads are 16KB-strided; tiles are staged into
// LDS as coalesced 512B rows using double-buffered async global->LDS DMA
// (GLOBAL_LOAD_ASYNC_TO_LDS_B128 + s_wait_asynccnt): the next tile's copy
// overlaps the current tile's 64 recurrence steps.
// ---------------------------------------------------------------------------

#define B_TOTAL   4096
#define T_LEN     2048
#define T_BLK     64
#define ROWF      132            // floats per LDS row: 128 data + 4 pad (bank spread)
#define WAVES_PB  2              // 2 waves/block so double-buffered LDS fits
#define TILE_B    16

typedef __attribute__((ext_vector_type(2))) float v2f;
typedef __attribute__((ext_vector_type(8))) float v8f;

#if __has_builtin(__builtin_amdgcn_global_load_async_to_lds_b128) && \
    __has_builtin(__builtin_amdgcn_s_wait_asynccnt)
#define USE_ASYNC 1
#else
#define USE_ASYNC 0
#endif

// Exact parameter types per clang's diagnostic:
//   arg0: v4i __attribute__((address_space(1)))*   (global source)
//   arg1: v4i __attribute__((address_space(3)))*   (LDS destination)
typedef __attribute__((__vector_size__(4 * sizeof(int)))) int v4i;
typedef __attribute__((address_space(1))) v4i* gv4i_ptr;
typedef __attribute__((address_space(3))) v4i* lv4i_ptr;

__device__ __forceinline__ float fast_tanh(float x) {
#if __has_builtin(__builtin_amdgcn_tanhf)
  return __builtin_amdgcn_tanhf(x);                    // v_tanh_f32, 1 TRANS op
#else
  float t = __builtin_amdgcn_exp2f(x * 2.88539008f);
  return 1.0f - 2.0f * __builtin_amdgcn_rcpf(t + 1.0f);
#endif
}

__device__ __forceinline__ float swap16(float x) {
  // ds_swizzle_b32 group-of-32: xor=0x10, or=0, and=0x1f -> lane n <-> n+16
  return __int_as_float(__builtin_amdgcn_ds_swizzle(__float_as_int(x), 0x401f));
}

// Stage x[base..base+15, t0..t0+T_BLK-1, 0:2] into one LDS buffer.
// Each batch row is 512B contiguous in global memory: 32 lanes x 16B per row.
__device__ __forceinline__ void stage_tile(const float* __restrict__ x,
                                           float* dst, int base, int t0,
                                           int lane) {
#if USE_ASYNC
  #pragma unroll
  for (int i = 0; i < TILE_B; ++i) {
    const float* g = x + ((size_t)(base + i) * T_LEN + t0) * 2 + lane * 4;
    float* l = dst + i * ROWF + lane * 4;
    gv4i_ptr gp = (gv4i_ptr)(uintptr_t)g;
    // generic LDS address = {shared aperture, ds offset}: low 32 bits = DS addr
    lv4i_ptr lp = (lv4i_ptr)(uintptr_t)(uint32_t)(uintptr_t)l;
    __builtin_amdgcn_global_load_async_to_lds_b128(gp, lp, 0, 0);
  }
#else
  #pragma unroll
  for (int i = 0; i < TILE_B; ++i) {
    const float4 vv =
        *(const float4*)(x + ((size_t)(base + i) * T_LEN + t0) * 2 + lane * 4);
    *(float4*)(dst + i * ROWF + lane * 4) = vv;
  }
#endif
}

__device__ __forceinline__ void wait_stage() {
#if USE_ASYNC
  __builtin_amdgcn_s_wait_asynccnt(0);
  asm volatile("" ::: "memory");   // keep LDS reads ordered after the wait
#endif
  // sync path: DS ops are in-order within a wave; compiler inserts dscnt waits
}

__global__ void __launch_bounds__(WAVES_PB * 32)
lstm_attn_fused(const float* __restrict__ x,
                const float* __restrict__ W_ih,
                const float* __restrict__ W_hh,
                const float* __restrict__ b_ih,
                const float* __restrict__ b_hh,
                const float* __restrict__ fc_w,
                const float* __restrict__ fc_bp,
                float* __restrict__ out)
{
  __shared__ float lds[WAVES_PB * 2 * TILE_B * ROWF];  // double buffered

  const int tid  = threadIdx.x;
  const int wave = tid >> 5;
  const int lane = tid & 31;
  const int n    = lane & 15;     // batch column within tile (D col, A row)
  const int up   = lane >> 4;     // 0: gate rows 0-7 (i,f); 1: rows 8-15 (g,o)
  const int base = (blockIdx.x * WAVES_PB + wave) * TILE_B;
  float* buf0 = lds + wave * 2 * TILE_B * ROWF;
  float* buf1 = buf0 + TILE_B * ROWF;

  // ---- per-lane constant weights -----------------------------------------
  float wih0[8], wih1[8], bias[8];
  #pragma unroll
  for (int v = 0; v < 8; ++v) {
    const int r = up * 8 + v;            // gate row this lane owns in C/D layout
    wih0[v] = W_ih[r * 2 + 0];
    wih1[v] = W_ih[r * 2 + 1];
    bias[v] = b_ih[r] + b_hh[r];
  }
  // A = W_hh (16x4) f32: lane m holds K=0,1; lane m+16 holds K=2,3
  v2f amat;
  amat.x = W_hh[n * 4 + up * 2 + 0];
  amat.y = W_hh[n * 4 + up * 2 + 1];

  const float fw0 = fc_w[0], fw1 = fc_w[1], fw2 = fc_w[2], fw3 = fc_w[3];
  const float fw4 = fc_w[4], fw5 = fc_w[5], fw6 = fc_w[6], fw7 = fc_w[7];
  const float fcb = fc_bp[0];

  // e1 path: low half sigmoid(i) = 0.5*tanh(0.5x)+0.5 ; high half tanh(g)
  const float s1 = up ? 1.0f : 0.5f;
  const float m1 = up ? 1.0f : 0.5f;
  const float a1 = up ? 0.0f : 0.5f;

  float q0 = 0.f, q1 = 0.f, q2 = 0.f, q3 = 0.f;
  float result = 0.f;

  #pragma unroll
  for (int pass = 0; pass < 2; ++pass) {
    float h[4] = {0.f, 0.f, 0.f, 0.f};
    float c[4] = {0.f, 0.f, 0.f, 0.f};
    float mx = -3.0e38f, S = 0.f;
    float A0 = 0.f, A1 = 0.f, A2 = 0.f, A3 = 0.f;  // softmax-weighted h accum

    stage_tile(x, buf0, base, 0, lane);
    wait_stage();

    for (int t0 = 0; t0 < T_LEN; t0 += T_BLK) {
      const int cur = (t0 / T_BLK) & 1;
      float* cbuf = cur ? buf1 : buf0;
      float* nbuf = cur ? buf0 : buf1;

      // Kick off the next tile's global->LDS DMA before computing this one.
      if (t0 + T_BLK < T_LEN)
        stage_tile(x, nbuf, base, t0 + T_BLK, lane);

      #pragma unroll 4
      for (int tl = 0; tl < T_BLK; ++tl) {
        const float2 xv = *(const float2*)(cbuf + n * ROWF + tl * 2);

        // C = x @ W_ih^T + (b_ih + b_hh), in 16x16 f32 D layout
        v8f acc;
        #pragma unroll
        for (int v = 0; v < 8; ++v)
          acc[v] = __builtin_fmaf(wih1[v], xv.y,
                   __builtin_fmaf(wih0[v], xv.x, bias[v]));

        // B = h^T (4 x 16 batch): v0 = {h0 | h2}, v1 = {h1 | h3}
        v2f bm;
        bm.x = up ? h[2] : h[0];
        bm.y = up ? h[3] : h[1];

        // gates^T = W_hh @ h^T + C    (K = 4 = HIDDEN, exact f32)
        acc = __builtin_amdgcn_wmma_f32_16x16x4_f32(false, amat, false, bm,
                                                    (short)0, acc, false, false);

        float e1[4], e2[4], sw1[4], sw2[4];
        #pragma unroll
        for (int j = 0; j < 4; ++j) {
          e1[j] = __builtin_fmaf(m1, fast_tanh(acc[j] * s1), a1);           // sig(i)|tanh(g)
          e2[j] = __builtin_fmaf(0.5f, fast_tanh(acc[4 + j] * 0.5f), 0.5f); // sig(f)|sig(o)
        }
        #pragma unroll
        for (int j = 0; j < 4; ++j) {
          sw1[j] = swap16(e1[j]);   // low gets tanh(g), high gets sig(i)
          sw2[j] = swap16(e2[j]);   // low gets sig(o),  high gets sig(f)
        }
        #pragma unroll
        for (int j = 0; j < 4; ++j) {
          const float F  = up ? sw2[j] : e2[j];     // sig(f)
          const float O  = up ? e2[j] : sw2[j];     // sig(o)
          const float ig = e1[j] * sw1[j];          // sig(i)*tanh(g), both halves
          c[j] = __builtin_fmaf(F, c[j], ig);
          h[j] = O * fast_tanh(c[j]);
        }

        if (pass == 1) {
          // online softmax attention, query q = h_T
          const float l  = __builtin_fmaf(h[3], q3, __builtin_fmaf(h[2], q2,
                           __builtin_fmaf(h[1], q1, h[0] * q0)));
          const float mn = fmaxf(mx, l);
          const float sc = __builtin_amdgcn_exp2f((mx - mn) * 1.44269504f);
          const float p  = __builtin_amdgcn_exp2f((l  - mn) * 1.44269504f);
          S  = __builtin_fmaf(S,  sc, p);
          A0 = __builtin_fmaf(A0, sc, p * h[0]);
          A1 = __builtin_fmaf(A1, sc, p * h[1]);
          A2 = __builtin_fmaf(A2, sc, p * h[2]);
          A3 = __builtin_fmaf(A3, sc, p * h[3]);
          mx = mn;
        }
      }

      wait_stage();   // next buffer's DMA must be complete before we swap
    }

    if (pass == 0) {
      q0 = h[0]; q1 = h[1]; q2 = h[2]; q3 = h[3];
    } else {
      const float inv = __builtin_amdgcn_rcpf(S);
      float r = __builtin_fmaf(fw0, q0, fcb);
      r = __builtin_fmaf(fw1, q1, r);
      r = __builtin_fmaf(fw2, q2, r);
      r = __builtin_fmaf(fw3, q3, r);
      r = __builtin_fmaf(fw4, A0 * inv, r);
      r = __builtin_fmaf(fw5, A1 * inv, r);
      r = __builtin_fmaf(fw6, A2 * inv, r);
      r = __builtin_fmaf(fw7, A3 * inv, r);
      result = r;
    }
  }

  if (up == 0) out[base + n] = result;
}

extern "C" void kernel_launch(void* const* d_in, const int* in_sizes, int n_in,
                              void* d_out, int out_size, void* d_ws, size_t ws_size,
                              hipStream_t stream) {
  (void)in_sizes; (void)n_in; (void)d_ws; (void)ws_size; (void)out_size;
  const float* x    = (const float*)d_in[0];
  const float* W_ih = (const float*)d_in[1];
  const float* W_hh = (const float*)d_in[2];
  const float* b_ih = (const float*)d_in[3];
  const float* b_hh = (const float*)d_in[4];
  const float* fc_w = (const float*)d_in[5];
  const float* fc_b = (const float*)d_in[6];
  float* out = (float*)d_out;

  const int blocks = B_TOTAL / (WAVES_PB * TILE_B);   // 128 blocks x 64 threads
  lstm_attn_fused<<<blocks, WAVES_PB * 32, 0, stream>>>(
      x, W_ih, W_hh, b_ih, b_hh, fc_w, fc_b, out);
}